// GNNModel_32100585570932
// MI455X (gfx1250) — compile-verified
//
#include <hip/hip_runtime.h>
#include <math.h>

#define N_NODES   100000
#define N_EDGES   6400000
#define F_IN      128
#define F_HID     16
#define N_CLASSES 10

typedef __attribute__((ext_vector_type(2))) float v2f;
typedef __attribute__((ext_vector_type(8))) float v8f;

// ---------------- degree / normalization ----------------

__global__ void k_init_deg(float* __restrict__ deg) {
    int i = blockIdx.x * blockDim.x + threadIdx.x;
    if (i < N_NODES) deg[i] = 1.0f;  // self-loop contributes 1
}

__global__ void k_accum_deg(const int* __restrict__ dst, float* __restrict__ deg) {
    int e = blockIdx.x * blockDim.x + threadIdx.x;
    if (e < N_EDGES) atomicAdd(&deg[dst[e]], 1.0f);
}

__global__ void k_dinv(float* __restrict__ deg) {
    int i = blockIdx.x * blockDim.x + threadIdx.x;
    if (i < N_NODES) deg[i] = rsqrtf(deg[i]);  // deg >= 1 always
}

// ---------------- GEMM 1: h1 = x @ W1  (100000x128 @ 128x16), fp32 WMMA ----------------
// One wave computes one 16x16 output tile; K=128 -> 32 x V_WMMA_F32_16X16X4_F32.
// A 16x4 f32 layout: lanes 0-15 hold M=lane, K={0,1}; lanes 16-31 hold M=lane-16, K={2,3}.
// B 4x16 f32 layout: lanes 0-15 hold N=lane, K={0,1}; lanes 16-31 hold N=lane-16, K={2,3}.
// C/D: VGPR r -> (M=r, N=lane) lanes 0-15; (M=r+8, N=lane-16) lanes 16-31.

__global__ void k_gemm1(const float* __restrict__ x, const float* __restrict__ W1,
                        float* __restrict__ h1) {
    const int lane = threadIdx.x & 31;
    const int wave = threadIdx.x >> 5;
    const int tile = blockIdx.x * (blockDim.x >> 5) + wave;
    if (tile >= N_NODES / 16) return;  // wave-uniform: EXEC all-1 for WMMA

    const int m0   = tile * 16;
    const int col  = lane & 15;
    const int hi   = lane >> 4;   // 0 or 1
    const int koff = hi * 2;      // K sub-offset for this half-wave

    const float* xrow = x + (size_t)(m0 + col) * F_IN;  // A row for this lane (M = lane&15)

    v8f c = {0.f, 0.f, 0.f, 0.f, 0.f, 0.f, 0.f, 0.f};
    #pragma unroll 4
    for (int k0 = 0; k0 < F_IN; k0 += 4) {
        v2f a = *(const v2f*)(xrow + k0 + koff);  // contiguous pair, 8B aligned
        v2f b;
        b.x = W1[(k0 + koff + 0) * F_HID + col];
        b.y = W1[(k0 + koff + 1) * F_HID + col];
        c = __builtin_amdgcn_wmma_f32_16x16x4_f32(false, a, false, b,
                                                  (short)0, c, false, false);
    }

    float* out = h1 + (size_t)m0 * F_HID;
    const int mh = hi * 8;
    #pragma unroll
    for (int r = 0; r < 8; ++r)
        out[(size_t)(r + mh) * F_HID + col] = c[r];
}

// ---------------- scatter-add layer 1: agg1[dst] += h1[src] * dinv[src]*dinv[dst] -------
// thread -> (edge, feature); 16 lanes per edge read one 64B segment of h1 (coalesced).
// Edges [N_EDGES, N_EDGES+N_NODES) are the self-loops (src = dst = node).

__global__ void k_scatter1(const int* __restrict__ ei, const float* __restrict__ dinv,
                           const float* __restrict__ h1, float* __restrict__ agg1) {
    int t = blockIdx.x * blockDim.x + threadIdx.x;   // total = (E+N)*16 = 104e6 < 2^31
    const int total = (N_EDGES + N_NODES) * 16;
    if (t >= total) return;
    int f = t & 15;
    int e = t >> 4;
    int s, d;
    if (e < N_EDGES) { s = ei[e]; d = ei[N_EDGES + e]; }
    else             { s = d = e - N_EDGES; }
    float norm = dinv[s] * dinv[d];
    atomicAdd(&agg1[(size_t)d * F_HID + f], h1[(size_t)s * F_HID + f] * norm);
}

// ---------------- GEMM 2: h2 = relu(agg1 + b1) @ W2 (bias+relu fused into A load) -------

__global__ void k_gemm2(const float* __restrict__ agg1, const float* __restrict__ b1,
                        const float* __restrict__ W2, float* __restrict__ h2) {
    const int lane = threadIdx.x & 31;
    const int wave = threadIdx.x >> 5;
    const int tile = blockIdx.x * (blockDim.x >> 5) + wave;
    if (tile >= N_NODES / 16) return;

    const int m0   = tile * 16;
    const int col  = lane & 15;
    const int hi   = lane >> 4;
    const int koff = hi * 2;

    const float* arow = agg1 + (size_t)(m0 + col) * F_HID;
    const bool colok = (col < N_CLASSES);

    v8f c = {0.f, 0.f, 0.f, 0.f, 0.f, 0.f, 0.f, 0.f};
    #pragma unroll
    for (int k0 = 0; k0 < F_HID; k0 += 4) {
        const int ka = k0 + koff;
        v2f a;
        a.x = fmaxf(arow[ka + 0] + b1[ka + 0], 0.0f);
        a.y = fmaxf(arow[ka + 1] + b1[ka + 1], 0.0f);
        v2f b;
        b.x = colok ? W2[(ka + 0) * N_CLASSES + col] : 0.0f;
        b.y = colok ? W2[(ka + 1) * N_CLASSES + col] : 0.0f;
        c = __builtin_amdgcn_wmma_f32_16x16x4_f32(false, a, false, b,
                                                  (short)0, c, false, false);
    }

    float* out = h2 + (size_t)m0 * F_HID;   // stride 16, cols 10..15 are exact zeros
    const int mh = hi * 8;
    #pragma unroll
    for (int r = 0; r < 8; ++r)
        out[(size_t)(r + mh) * F_HID + col] = c[r];
}

// ---------------- scatter-add layer 2 (10 live features, stride-16 storage) -------------

__global__ void k_scatter2(const int* __restrict__ ei, const float* __restrict__ dinv,
                           const float* __restrict__ h2, float* __restrict__ agg2) {
    int t = blockIdx.x * blockDim.x + threadIdx.x;
    const int total = (N_EDGES + N_NODES) * 16;
    if (t >= total) return;
    int f = t & 15;
    if (f >= N_CLASSES) return;
    int e = t >> 4;
    int s, d;
    if (e < N_EDGES) { s = ei[e]; d = ei[N_EDGES + e]; }
    else             { s = d = e - N_EDGES; }
    float norm = dinv[s] * dinv[d];
    atomicAdd(&agg2[(size_t)d * 16 + f], h2[(size_t)s * 16 + f] * norm);
}

// ---------------- bias + log_softmax -> d_out ----------------

__global__ void k_finalize(const float* __restrict__ agg2, const float* __restrict__ b2,
                           float* __restrict__ out) {
    int i = blockIdx.x * blockDim.x + threadIdx.x;
    if (i >= N_NODES) return;
    float z[N_CLASSES];
    float m = -INFINITY;
    #pragma unroll
    for (int cidx = 0; cidx < N_CLASSES; ++cidx) {
        z[cidx] = agg2[(size_t)i * 16 + cidx] + b2[cidx];
        m = fmaxf(m, z[cidx]);
    }
    float s = 0.0f;
    #pragma unroll
    for (int cidx = 0; cidx < N_CLASSES; ++cidx) s += __expf(z[cidx] - m);
    const float ls = m + __logf(s);
    #pragma unroll
    for (int cidx = 0; cidx < N_CLASSES; ++cidx)
        out[(size_t)i * N_CLASSES + cidx] = z[cidx] - ls;
}

// ---------------- host launch ----------------

extern "C" void kernel_launch(void* const* d_in, const int* in_sizes, int n_in,
                              void* d_out, int out_size, void* d_ws, size_t ws_size,
                              hipStream_t stream) {
    (void)in_sizes; (void)n_in; (void)out_size; (void)ws_size;
    const float* x  = (const float*)d_in[0];
    const int*   ei = (const int*)  d_in[1];   // [2, E]: src row then dst row
    const float* W1 = (const float*)d_in[2];
    const float* b1 = (const float*)d_in[3];
    const float* W2 = (const float*)d_in[4];
    const float* b2 = (const float*)d_in[5];
    float* out = (float*)d_out;

    // workspace layout (floats): dinv[N] | h1/agg2[16N] | agg1[16N] | h2[16N]  = 49N
    float* ws   = (float*)d_ws;
    float* dinv = ws;
    float* h1   = ws + N_NODES;                 // reused as agg2 after scatter1
    float* agg2 = h1;
    float* agg1 = ws + (size_t)N_NODES * 17;
    float* h2   = ws + (size_t)N_NODES * 33;

    const int ntiles      = N_NODES / 16;                 // 6250, exact
    const int gemm_blocks = (ntiles + 7) / 8;             // 8 waves of 32 per block
    const int scat_blocks = ((N_EDGES + N_NODES) * 16 + 255) / 256;  // 406250

    k_init_deg <<<(N_NODES + 255) / 256, 256, 0, stream>>>(dinv);
    k_accum_deg<<<(N_EDGES + 255) / 256, 256, 0, stream>>>(ei + N_EDGES, dinv);
    k_dinv     <<<(N_NODES + 255) / 256, 256, 0, stream>>>(dinv);

    hipMemsetAsync(agg1, 0, (size_t)N_NODES * 16 * sizeof(float), stream);
    k_gemm1   <<<gemm_blocks, 256, 0, stream>>>(x, W1, h1);
    k_scatter1<<<scat_blocks, 256, 0, stream>>>(ei, dinv, h1, agg1);

    // h1 is dead now; its buffer becomes agg2 (stream-ordered, safe)
    hipMemsetAsync(agg2, 0, (size_t)N_NODES * 16 * sizeof(float), stream);
    k_gemm2   <<<gemm_blocks, 256, 0, stream>>>(agg1, b1, W2, h2);
    k_scatter2<<<scat_blocks, 256, 0, stream>>>(ei, dinv, h2, agg2);

    k_finalize<<<(N_NODES + 255) / 256, 256, 0, stream>>>(agg2, b2, out);
}